// GCNLayer_55224689492697
// MI455X (gfx1250) — compile-verified
//
#include <hip/hip_runtime.h>

typedef __attribute__((ext_vector_type(2))) float v2f;
typedef __attribute__((ext_vector_type(8))) float v8f;

#define DF 128
#define BN_EPS 1e-5f

// ---------------------------------------------------------------------------
// Kernel 1: xw = x @ W  via V_WMMA_F32_16X16X4_F32.
// Block = 256 threads = 8 waves. Block owns a 16-row strip of x (staged in
// LDS, reused by all 8 waves); wave w computes the 16x16 output tile at
// columns [16w, 16w+16). W (64 KB) stays in global and is L2-resident.
// ---------------------------------------------------------------------------
__global__ __launch_bounds__(256) void gemm_xw_wmma(
    const float* __restrict__ x, const float* __restrict__ W,
    float* __restrict__ xw, int N) {
  __shared__ float sX[16 * DF];
  const int row0 = blockIdx.x * 16;
  const int tid = threadIdx.x;

  // cooperative load of the 16x128 strip (guarded for a ragged last block)
  for (int i = tid; i < 16 * DF; i += 256) {
    const int r = row0 + (i >> 7);
    sX[i] = (r < N) ? x[(size_t)r * DF + (i & (DF - 1))] : 0.0f;
  }
  __syncthreads();

  const int wave = tid >> 5;          // 0..7 -> 16-column tile
  const int lane = tid & 31;
  const int col0 = wave * 16;
  const int m    = lane & 15;         // A: M index / B: N index
  const int koff = (lane >> 4) << 1;  // lanes 0-15 -> K+0/K+1, 16-31 -> K+2/K+3

  v8f acc = {};
  for (int k0 = 0; k0 < DF; k0 += 4) {
    // A fragment: 16x4 fp32, row M striped in-lane (ISA 7.12.2)
    v2f a = *(const v2f*)&sX[m * DF + k0 + koff];
    // B fragment: 4x16 fp32, row K striped across lanes
    v2f b;
    b.x = W[(size_t)(k0 + koff)     * DF + col0 + m];
    b.y = W[(size_t)(k0 + koff + 1) * DF + col0 + m];
    acc = __builtin_amdgcn_wmma_f32_16x16x4_f32(
        /*neg_a=*/false, a, /*neg_b=*/false, b,
        /*c_mod=*/(short)0, acc, /*reuse_a=*/false, /*reuse_b=*/false);
  }

  // C/D layout: VGPR r -> M = r (lanes 0-15) or r+8 (lanes 16-31); N = lane&15
  const int mbase = row0 + ((lane >> 4) << 3);
#pragma unroll
  for (int r = 0; r < 8; ++r) {
    const int row = mbase + r;
    if (row < N) xw[(size_t)row * DF + col0 + m] = acc[r];
  }
}

// ---------------------------------------------------------------------------
// Kernel 2: deg[i] = 1.0 (self-loop); zero the BN accumulators.
// ---------------------------------------------------------------------------
__global__ void init_deg_stats(float* __restrict__ deg, float* __restrict__ sum,
                               float* __restrict__ sumsq, int N) {
  const int i = blockIdx.x * blockDim.x + threadIdx.x;
  if (i < N) deg[i] = 1.0f;
  if (i < DF) { sum[i] = 0.0f; sumsq[i] = 0.0f; }
}

// ---------------------------------------------------------------------------
// Kernel 3: in-degree accumulation over edge destinations.
// ---------------------------------------------------------------------------
__global__ void deg_accum(const long long* __restrict__ ei,
                          float* __restrict__ deg, int nEdges) {
  const int e = blockIdx.x * blockDim.x + threadIdx.x;
  if (e < nEdges) atomicAdd(&deg[(int)ei[nEdges + e]], 1.0f);
}

// ---------------------------------------------------------------------------
// Kernel 4: deg -> rsqrt(deg) in place (deg >= 1 always, so no zero guard).
// ---------------------------------------------------------------------------
__global__ void deg_to_dinv(float* __restrict__ deg, int N) {
  const int i = blockIdx.x * blockDim.x + threadIdx.x;
  if (i < N) deg[i] = rsqrtf(deg[i]);
}

// ---------------------------------------------------------------------------
// Kernel 5: h = b + xw * dinv^2   (self-loop message; h lives in d_out)
// ---------------------------------------------------------------------------
__global__ __launch_bounds__(256) void h_init(
    const float* __restrict__ xw, const float* __restrict__ dinv,
    const float* __restrict__ b, float* __restrict__ h, int total) {
  const int i = blockIdx.x * blockDim.x + threadIdx.x;
  if (i >= total) return;
  const int node = i >> 7;
  const int c = i & (DF - 1);
  const float di = dinv[node];
  h[i] = b[c] + xw[i] * di * di;
}

// ---------------------------------------------------------------------------
// Kernel 6: per-edge gather + scale + scatter-add. One wave32 per edge; each
// lane moves a float4 slice of the 128-wide message (coalesced 512B/edge).
// ---------------------------------------------------------------------------
__global__ __launch_bounds__(256) void scatter_edges(
    const long long* __restrict__ ei, const float* __restrict__ xw,
    const float* __restrict__ dinv, float* __restrict__ h, int nEdges) {
  const int lane = threadIdx.x & 31;
  const int e = blockIdx.x * (blockDim.x >> 5) + (threadIdx.x >> 5);
  if (e >= nEdges) return;
  const int src = (int)ei[e];
  const int dst = (int)ei[nEdges + e];
  const float norm = dinv[src] * dinv[dst];
  const float4 v = *(const float4*)&xw[(size_t)src * DF + lane * 4];
  float* hd = &h[(size_t)dst * DF + lane * 4];
  atomicAdd(hd + 0, v.x * norm);
  atomicAdd(hd + 1, v.y * norm);
  atomicAdd(hd + 2, v.z * norm);
  atomicAdd(hd + 3, v.w * norm);
}

// ---------------------------------------------------------------------------
// Kernel 7: column-wise sum / sum-of-squares partial reduction.
// blockDim = 128 (thread = column); grid strides over rows.
// ---------------------------------------------------------------------------
__global__ __launch_bounds__(128) void bn_stats(
    const float* __restrict__ h, float* __restrict__ sum,
    float* __restrict__ sumsq, int N) {
  const int col = threadIdx.x;
  float s = 0.0f, ss = 0.0f;
  for (int i = blockIdx.x; i < N; i += gridDim.x) {
    const float v = h[(size_t)i * DF + col];
    s += v;
    ss += v * v;
  }
  atomicAdd(&sum[col], s);
  atomicAdd(&sumsq[col], ss);
}

// ---------------------------------------------------------------------------
// Kernel 8: finalize BN affine:  scale = g*rsqrt(var+eps), shift = b - mean*scale
// ---------------------------------------------------------------------------
__global__ void bn_finalize(const float* __restrict__ sum,
                            const float* __restrict__ sumsq,
                            const float* __restrict__ gamma,
                            const float* __restrict__ beta,
                            float* __restrict__ scale,
                            float* __restrict__ shift, int N) {
  const int c = threadIdx.x;
  if (c < DF) {
    const float invN = 1.0f / (float)N;
    const float mean = sum[c] * invN;
    const float var = sumsq[c] * invN - mean * mean;
    const float sc = gamma[c] * rsqrtf(var + BN_EPS);
    scale[c] = sc;
    shift[c] = beta[c] - mean * sc;
  }
}

// ---------------------------------------------------------------------------
// Kernel 9: h = x + relu(h*scale + shift), in place on d_out.
// ---------------------------------------------------------------------------
__global__ __launch_bounds__(256) void bn_act_res(
    const float* __restrict__ x, const float* __restrict__ scale,
    const float* __restrict__ shift, float* __restrict__ h, int total) {
  const int i = blockIdx.x * blockDim.x + threadIdx.x;
  if (i >= total) return;
  const int c = i & (DF - 1);
  float v = fmaf(h[i], scale[c], shift[c]);
  v = v > 0.0f ? v : 0.0f;
  h[i] = x[i] + v;
}

// ---------------------------------------------------------------------------
extern "C" void kernel_launch(void* const* d_in, const int* in_sizes, int n_in,
                              void* d_out, int out_size, void* d_ws,
                              size_t ws_size, hipStream_t stream) {
  const float* x      = (const float*)d_in[0];
  const float* W      = (const float*)d_in[1];
  const float* b      = (const float*)d_in[2];
  const float* gamma  = (const float*)d_in[3];
  const float* beta   = (const float*)d_in[4];
  const long long* ei = (const long long*)d_in[5];  // int64 per reference

  const int N = in_sizes[0] / DF;   // 50000
  const int E = in_sizes[5] / 2;    // 640000
  const int total = N * DF;

  float* h = (float*)d_out;         // aggregation buffer lives in d_out
  float* ws = (float*)d_ws;
  float* xw    = ws;                       // N*DF floats (~25.6 MB)
  float* dinv  = ws + (size_t)N * DF;      // N floats (deg, then rsqrt in place)
  float* sum   = dinv + N;                 // DF
  float* sumsq = sum + DF;                 // DF
  float* scale = sumsq + DF;               // DF
  float* shift = scale + DF;               // DF

  // 1) GEMM via fp32 WMMA
  gemm_xw_wmma<<<(N + 15) / 16, 256, 0, stream>>>(x, W, xw, N);
  // 2) deg = 1 (self loops), zero BN accumulators
  init_deg_stats<<<(N + 255) / 256, 256, 0, stream>>>(dinv, sum, sumsq, N);
  // 3) degree accumulation
  deg_accum<<<(E + 255) / 256, 256, 0, stream>>>(ei, dinv, E);
  // 4) dinv = rsqrt(deg)
  deg_to_dinv<<<(N + 255) / 256, 256, 0, stream>>>(dinv, N);
  // 5) h = b + self-loop message
  h_init<<<(total + 255) / 256, 256, 0, stream>>>(xw, dinv, b, h, total);
  // 6) edge scatter (bandwidth-dominant step)
  scatter_edges<<<(E + 7) / 8, 256, 0, stream>>>(ei, xw, dinv, h, E);
  // 7) BN statistics
  bn_stats<<<512, 128, 0, stream>>>(h, sum, sumsq, N);
  // 8) finalize affine
  bn_finalize<<<1, 128, 0, stream>>>(sum, sumsq, gamma, beta, scale, shift, N);
  // 9) normalize + relu + residual
  bn_act_res<<<(total + 255) / 256, 256, 0, stream>>>(x, scale, shift, h, total);
}